// AutoRelGraphConvolution_17076789969303
// MI455X (gfx1250) — compile-verified
//
#include <hip/hip_runtime.h>
#include <cstdint>
#include <cstddef>

#define N_ENT   100000
#define N_REL   500
#define DIMS    128
#define NEDGE   1000000
#define ALPHA_F 0.5f
#define BETA_F  0.5f
#define EPS_F   1e-12f

#define EDGE_BLOCKS  256
#define EDGE_THREADS 1024
#define WAVES_PER_BLOCK (EDGE_THREADS / 32)
#define CHUNK 1024

#if __has_builtin(__builtin_amdgcn_global_load_async_to_lds_b32) && __has_builtin(__builtin_amdgcn_s_wait_asynccnt)
#define HAVE_ASYNC_LDS 1
typedef __attribute__((address_space(1))) int as1_int;   // global ("__device__") int*
typedef __attribute__((address_space(3))) int as3_int;   // LDS ("__shared__") int*
#endif

static __device__ __forceinline__ void gatomic_add(float* p, float v) {
  (void)__hip_atomic_fetch_add(p, v, __ATOMIC_RELAXED, __HIP_MEMORY_SCOPE_AGENT);
}
static __device__ __forceinline__ void latomic_add(float* p, float v) {
  (void)__hip_atomic_fetch_add(p, v, __ATOMIC_RELAXED, __HIP_MEMORY_SCOPE_WORKGROUP);
}

// ---------------------------------------------------------------------------
// Kernel 1: pairwise L2 normalization of rel_emb -> rel_n (in workspace)
// ---------------------------------------------------------------------------
__global__ void relnorm_kernel(const float* __restrict__ rel_emb,
                               float* __restrict__ rel_n) {
  int i = blockIdx.x * blockDim.x + threadIdx.x;   // complex-pair index
  if (i >= N_REL * DIMS / 2) return;
  float rr = rel_emb[2 * i + 0];
  float ri = rel_emb[2 * i + 1];
  float inv = 1.0f / sqrtf(fmaxf(rr * rr + ri * ri, EPS_F));
  rel_n[2 * i + 0] = rr * inv;
  rel_n[2 * i + 1] = ri * inv;
}

// ---------------------------------------------------------------------------
// Kernel 2: edge scatter-gather. One wave32 per edge; lane owns one float4
// (= 2 complex dims). rel gradients accumulate in a 256KB LDS table
// (possible only with CDNA5's 320KB LDS), flushed once per block.
// Edge indices are staged into LDS with CDNA5 async global->LDS loads.
// ---------------------------------------------------------------------------
__global__ void __launch_bounds__(EDGE_THREADS, 1)
edge_kernel(const float* __restrict__ ent_emb,
            const float* __restrict__ rel_n,
            const int*   __restrict__ head,
            const int*   __restrict__ relx,
            const int*   __restrict__ tail,
            float*       __restrict__ ent_msg,
            float*       __restrict__ rel_msg) {
  __shared__ float s_racc[N_REL * DIMS];  // 256,000 B relation-grad accumulator
  __shared__ int   s_h[CHUNK];
  __shared__ int   s_r[CHUNK];
  __shared__ int   s_t[CHUNK];

  const int tid  = threadIdx.x;
  const int lane = tid & 31;
  const int wave = tid >> 5;

  for (int i = tid; i < N_REL * DIMS; i += EDGE_THREADS) s_racc[i] = 0.0f;
  __syncthreads();

  const int epb = (NEDGE + EDGE_BLOCKS - 1) / EDGE_BLOCKS;
  const int e0  = blockIdx.x * epb;
  const int e1  = (e0 + epb < NEDGE) ? (e0 + epb) : NEDGE;

  for (int base = e0; base < e1; base += CHUNK) {
    const int n = ((e1 - base) < CHUNK) ? (e1 - base) : CHUNK;

    if (tid < n) {
#ifdef HAVE_ASYNC_LDS
      __builtin_amdgcn_global_load_async_to_lds_b32(
          (as1_int*)(head + base + tid), (as3_int*)&s_h[tid], 0, 0);
      __builtin_amdgcn_global_load_async_to_lds_b32(
          (as1_int*)(relx + base + tid), (as3_int*)&s_r[tid], 0, 0);
      __builtin_amdgcn_global_load_async_to_lds_b32(
          (as1_int*)(tail + base + tid), (as3_int*)&s_t[tid], 0, 0);
#else
      s_h[tid] = head[base + tid];
      s_r[tid] = relx[base + tid];
      s_t[tid] = tail[base + tid];
#endif
    }
#ifdef HAVE_ASYNC_LDS
    __builtin_amdgcn_s_wait_asynccnt(0);
#endif
    __syncthreads();

    for (int j = wave; j < n; j += WAVES_PER_BLOCK) {
      if (j + WAVES_PER_BLOCK < n) {
        // prefetch next edge's gather rows (global_prefetch_b8)
        __builtin_prefetch(ent_emb + (size_t)s_h[j + WAVES_PER_BLOCK] * DIMS + lane * 4, 0, 0);
        __builtin_prefetch(ent_emb + (size_t)s_t[j + WAVES_PER_BLOCK] * DIMS + lane * 4, 0, 0);
      }
      const int hi = s_h[j];
      const int ri = s_r[j];
      const int ti = s_t[j];

      const float4 h = ((const float4*)(ent_emb + (size_t)hi * DIMS))[lane];
      const float4 t = ((const float4*)(ent_emb + (size_t)ti * DIMS))[lane];
      const float4 r = ((const float4*)(rel_n   + (size_t)ri * DIMS))[lane];

      // diff = h*r - t (complex, pairs (x,y) and (z,w))
      const float dr0 = (h.x * r.x - h.y * r.y) - t.x;
      const float di0 = (h.x * r.y + h.y * r.x) - t.y;
      const float dr1 = (h.z * r.z - h.w * r.w) - t.z;
      const float di1 = (h.z * r.w + h.w * r.z) - t.w;

      // dL/dh = -2*conj(r)*diff ; dL/dr = -2*conj(h)*diff ; dL/dt = +2*diff
      const float ghx = -2.f * (dr0 * r.x + di0 * r.y);
      const float ghy =  2.f * (dr0 * r.y - di0 * r.x);
      const float ghz = -2.f * (dr1 * r.z + di1 * r.w);
      const float ghw =  2.f * (dr1 * r.w - di1 * r.z);

      const float grx = -2.f * (dr0 * h.x + di0 * h.y);
      const float gry =  2.f * (dr0 * h.y - di0 * h.x);
      const float grz = -2.f * (dr1 * h.z + di1 * h.w);
      const float grw =  2.f * (dr1 * h.w - di1 * h.z);

      float* mh = ent_msg + (size_t)hi * DIMS + lane * 4;
      float* mt = ent_msg + (size_t)ti * DIMS + lane * 4;
      float* mr = s_racc  + ri * DIMS + lane * 4;

      gatomic_add(mh + 0, ghx);
      gatomic_add(mh + 1, ghy);
      gatomic_add(mh + 2, ghz);
      gatomic_add(mh + 3, ghw);

      gatomic_add(mt + 0, 2.f * dr0);
      gatomic_add(mt + 1, 2.f * di0);
      gatomic_add(mt + 2, 2.f * dr1);
      gatomic_add(mt + 3, 2.f * di1);

      latomic_add(mr + 0, grx);
      latomic_add(mr + 1, gry);
      latomic_add(mr + 2, grz);
      latomic_add(mr + 3, grw);
    }
    __syncthreads();  // protect index staging buffers before next chunk
  }

  // flush block-local relation gradients to global accumulator
  for (int i = tid; i < N_REL * DIMS; i += EDGE_THREADS)
    gatomic_add(&rel_msg[i], s_racc[i]);
}

// ---------------------------------------------------------------------------
// Kernel 3: out_ent = relu(ent_emb + ALPHA*ent_invsum*ent_msg), float4 wide
// ---------------------------------------------------------------------------
__global__ void fin_ent_kernel(const float* __restrict__ ent_emb,
                               const float* __restrict__ invsum,
                               const float* __restrict__ msg,
                               float* __restrict__ out) {
  int i = blockIdx.x * blockDim.x + threadIdx.x;   // float4 index
  if (i >= N_ENT * DIMS / 4) return;
  const int row = i >> 5;                          // 32 float4 per 128-dim row
  const float s = ALPHA_F * invsum[row];
  const float4 e = ((const float4*)ent_emb)[i];
  const float4 m = ((const float4*)msg)[i];
  float4 o;
  o.x = fmaxf(0.f, e.x + s * m.x);
  o.y = fmaxf(0.f, e.y + s * m.y);
  o.z = fmaxf(0.f, e.z + s * m.z);
  o.w = fmaxf(0.f, e.w + s * m.w);
  ((float4*)out)[i] = o;
}

// ---------------------------------------------------------------------------
// Kernel 4: out_rel = relu(rel_n + BETA*rel_invsum*rel_msg)
// ---------------------------------------------------------------------------
__global__ void fin_rel_kernel(const float* __restrict__ rel_n,
                               const float* __restrict__ invsum,
                               const float* __restrict__ msg,
                               float* __restrict__ out) {
  int i = blockIdx.x * blockDim.x + threadIdx.x;   // float4 index
  if (i >= N_REL * DIMS / 4) return;
  const int row = i >> 5;
  const float s = BETA_F * invsum[row];
  const float4 e = ((const float4*)rel_n)[i];
  const float4 m = ((const float4*)msg)[i];
  float4 o;
  o.x = fmaxf(0.f, e.x + s * m.x);
  o.y = fmaxf(0.f, e.y + s * m.y);
  o.z = fmaxf(0.f, e.z + s * m.z);
  o.w = fmaxf(0.f, e.w + s * m.w);
  ((float4*)out)[i] = o;
}

// ---------------------------------------------------------------------------
extern "C" void kernel_launch(void* const* d_in, const int* in_sizes, int n_in,
                              void* d_out, int out_size, void* d_ws, size_t ws_size,
                              hipStream_t stream) {
  const float* ent_emb = (const float*)d_in[0];
  const float* rel_emb = (const float*)d_in[1];
  const float* ent_inv = (const float*)d_in[2];
  const float* rel_inv = (const float*)d_in[3];
  const int*   head    = (const int*)d_in[4];
  const int*   relx    = (const int*)d_in[5];
  const int*   tail    = (const int*)d_in[6];

  // Workspace layout (floats): [rel_n 64000][ent_msg 12,800,000][rel_msg 64000]
  float* ws      = (float*)d_ws;
  float* rel_n   = ws;
  float* ent_msg = ws + (size_t)N_REL * DIMS;
  float* rel_msg = ent_msg + (size_t)N_ENT * DIMS;

  // zero the gradient accumulators (ent_msg + rel_msg are contiguous)
  (void)hipMemsetAsync(ent_msg, 0,
                       ((size_t)N_ENT * DIMS + (size_t)N_REL * DIMS) * sizeof(float),
                       stream);

  relnorm_kernel<<<(N_REL * DIMS / 2 + 255) / 256, 256, 0, stream>>>(rel_emb, rel_n);

  edge_kernel<<<EDGE_BLOCKS, EDGE_THREADS, 0, stream>>>(
      ent_emb, rel_n, head, relx, tail, ent_msg, rel_msg);

  float* out_ent = (float*)d_out;
  float* out_rel = out_ent + (size_t)N_ENT * DIMS;

  fin_ent_kernel<<<(N_ENT * DIMS / 4 + 255) / 256, 256, 0, stream>>>(
      ent_emb, ent_inv, ent_msg, out_ent);
  fin_rel_kernel<<<(N_REL * DIMS / 4 + 255) / 256, 256, 0, stream>>>(
      rel_n, rel_inv, rel_msg, out_rel);
}